// TtQwen2Attention_41008347743005
// MI455X (gfx1250) — compile-verified
//
#include <hip/hip_runtime.h>
#include <hip/hip_bf16.h>

// ---------------------------------------------------------------------------
// Qwen2-style GQA decode step, fp32, MI455X (gfx1250, wave32).
// Memory floor ~655MB @ 23.3TB/s => ~28us; KV streaming dominates.
// GEMMs use V_WMMA_F32_16X16X4_F32 (full f32 precision); attention core is
// bandwidth-bound flash-decode with non-temporal b128 loads.
// ---------------------------------------------------------------------------

typedef __attribute__((ext_vector_type(2))) float v2f;
typedef __attribute__((ext_vector_type(8))) float v8f;
typedef __attribute__((ext_vector_type(4))) float f4;

#define B_   32
#define D_   3584
#define NH_  28
#define NKV_ 4
#define G_   7
#define HD_  128
#define SW_  4096
#define NQKV_ 4608            // (NH + 2*NKV) * HD
#define KCH_ 4                // K-chunks for GEMM parallelism
#define PSTR_ 132             // per-(chunk,head) attention partial stride: m,l,pad2,acc[128]

// workspace offsets (floats), all multiples of 4 (16B aligned)
#define OFF_QKVP 0                                   // [4][32][4608]
#define OFF_QROT (OFF_QKVP + KCH_*B_*NQKV_)          // [32][28][128]
#define OFF_KNEW (OFF_QROT + B_*NH_*HD_)             // [32][4][128]
#define OFF_VNEW (OFF_KNEW + B_*NKV_*HD_)            // [32][4][128]
#define OFF_ATT  (OFF_VNEW + B_*NKV_*HD_)            // [32][4][32][7][132]
#define OFF_CTX  (OFF_ATT  + B_*NKV_*32*G_*PSTR_)    // [32][3584]
#define OFF_OUTP (OFF_CTX  + B_*D_)                  // [4][32][3584]

// ---------------------------------------------------------------------------
// GEMM: part[kc][32][N] = A[32][K](chunk kc) @ W[K][N], via WMMA f32 16x16x4.
// One wave per (N-tile, K-chunk); each wave owns both M-tiles (rows 0-15/16-31)
// so W is streamed exactly once. A-layout: lane L holds row L&15, K-pair
// 2*(L>>4); B-layout: lane L holds col L&15, K-rows {0,1} (lanes 0-15) or
// {2,3} (lanes 16-31), per the 16x16x4 VGPR striping in CDNA5 ISA.
// ---------------------------------------------------------------------------
__global__ __launch_bounds__(256) void gemm32_f32_wmma(
    const float* __restrict__ A, const float* __restrict__ W,
    float* __restrict__ part, int K, int N)
{
    const int lane = threadIdx.x & 31;
    const int gw   = blockIdx.x * 8 + (threadIdx.x >> 5);
    const int nTiles = N >> 4;
    if (gw >= nTiles * KCH_) return;
    const int nt = gw % nTiles;
    const int kc = gw / nTiles;
    const int chunkK = K / KCH_;
    const int kSteps = chunkK >> 2;

    const int rowA = lane & 15;
    const int kOff = (lane >> 4) << 1;            // 0 or 2
    const int col  = (nt << 4) + (lane & 15);

    v8f c0 = {}; v8f c1 = {};
    const float* a0p = A + (size_t)rowA * K + kc * chunkK + kOff;
    const float* a1p = a0p + (size_t)16 * K;
    const float* wp  = W + (size_t)(kc * chunkK + kOff) * N + col;

    for (int s = 0; s < kSteps; ++s) {
        v2f a0; a0.x = a0p[0]; a0.y = a0p[1];
        v2f a1; a1.x = a1p[0]; a1.y = a1p[1];
        v2f bb; bb.x = wp[0];  bb.y = wp[N];
        c0 = __builtin_amdgcn_wmma_f32_16x16x4_f32(false, a0, false, bb, (short)0, c0, false, false);
        c1 = __builtin_amdgcn_wmma_f32_16x16x4_f32(false, a1, false, bb, (short)0, c1, false, false);
        a0p += 4; a1p += 4; wp += (size_t)4 * N;
    }

    // C layout: VGPR j -> row j (lanes 0-15) / j+8 (lanes 16-31), col = lane&15
    const int rhalf = (lane >> 4) << 3;
    float* p0 = part + (size_t)kc * 32 * N + col;
    #pragma unroll
    for (int j = 0; j < 8; ++j) {
        p0[(size_t)(j + rhalf) * N]      = c0[j];
        p0[(size_t)(j + rhalf + 16) * N] = c1[j];
    }
}

// ---------------------------------------------------------------------------
// Reduce K-chunk partials + bias, apply RoPE (exact: rot_mat is block-diagonal
// 2x2, the zero entries contribute exact 0.0f), scale q by HD^-0.5, emit
// q_rot / k_new / v_new.
// ---------------------------------------------------------------------------
__global__ __launch_bounds__(256) void rope_assemble(
    const float* __restrict__ part, const float* __restrict__ bias,
    const float* __restrict__ rot,
    float* __restrict__ qrot, float* __restrict__ knew, float* __restrict__ vnew)
{
    const int tid = blockIdx.x * 256 + threadIdx.x;
    const int QP = B_ * NH_ * (HD_ / 2);     // 57344 q rotation pairs
    const int KP = B_ * NKV_ * (HD_ / 2);    // 8192 k rotation pairs
    const int VN = B_ * NKV_ * HD_;          // 16384 v elements

    auto S = [&](int b, int col) -> float {
        return bias[col]
             + part[(size_t)(0 * B_ + b) * NQKV_ + col]
             + part[(size_t)(1 * B_ + b) * NQKV_ + col]
             + part[(size_t)(2 * B_ + b) * NQKV_ + col]
             + part[(size_t)(3 * B_ + b) * NQKV_ + col];
    };

    if (tid < QP) {
        const int j = tid & 63, h = (tid >> 6) % NH_, b = tid / (64 * NH_);
        const int col = h * HD_ + 2 * j;
        const float x0 = S(b, col), x1 = S(b, col + 1);
        const float c = rot[(2 * j) * HD_ + 2 * j];
        const float s = rot[(2 * j) * HD_ + 2 * j + 1];
        const float scale = 0.08838834764831845f;     // 128^-0.5
        float* q = qrot + ((size_t)b * NH_ + h) * HD_ + 2 * j;
        q[0] = (x0 * c - x1 * s) * scale;
        q[1] = (x0 * s + x1 * c) * scale;
    } else if (tid < QP + KP) {
        const int t = tid - QP;
        const int j = t & 63, kv = (t >> 6) % NKV_, b = t / (64 * NKV_);
        const int col = NH_ * HD_ + kv * HD_ + 2 * j;
        const float x0 = S(b, col), x1 = S(b, col + 1);
        const float c = rot[(2 * j) * HD_ + 2 * j];
        const float s = rot[(2 * j) * HD_ + 2 * j + 1];
        float* k = knew + ((size_t)b * NKV_ + kv) * HD_ + 2 * j;
        k[0] = x0 * c - x1 * s;
        k[1] = x0 * s + x1 * c;
    } else if (tid < QP + KP + VN) {
        const int t = tid - QP - KP;
        const int d = t & 127, kv = (t >> 7) % NKV_, b = t / (HD_ * NKV_);
        const int col = (NH_ + NKV_) * HD_ + kv * HD_ + d;
        vnew[((size_t)b * NKV_ + kv) * HD_ + d] = S(b, col);
    }
}

// wave32 reductions
__device__ inline float waveMax(float v) {
    #pragma unroll
    for (int off = 16; off > 0; off >>= 1) v = fmaxf(v, __shfl_xor(v, off, 32));
    return v;
}
__device__ inline float waveSum(float v) {
    #pragma unroll
    for (int off = 16; off > 0; off >>= 1) v += __shfl_xor(v, off, 32);
    return v;
}

// ---------------------------------------------------------------------------
// Flash-decode partials. One wave per (b, kv, chunk-of-128-positions):
// 32*4*32 = 4096 waves, 8 waves/block sharing (b,kv) -> one LDS q copy.
// Phase A: lane = position, 7-head dots against q in LDS. Online softmax via
// wave reductions. Phase B: lane = 4 dims, p broadcast by shuffle, V accum.
// Ring slot (current_pos % SW) reads the fresh k/v (inputs stay immutable).
// ---------------------------------------------------------------------------
__global__ __launch_bounds__(256) void attn_partial(
    const float* __restrict__ cache_k, const float* __restrict__ cache_v,
    const float* __restrict__ qrot, const float* __restrict__ knew,
    const float* __restrict__ vnew, const int* __restrict__ curp,
    float* __restrict__ part)
{
    const int tid  = threadIdx.x;
    const int lane = tid & 31;
    const int gw   = blockIdx.x * 8 + (tid >> 5);      // 0..4095
    const int chunk = gw & 31;
    const int kv    = (gw >> 5) & 3;
    const int b     = gw >> 7;
    const int cur   = *curp;
    const int slot  = cur % SW_;

    __shared__ __align__(16) float qlds[G_ * HD_];
    const float* qbase = qrot + ((size_t)b * NH_ + kv * G_) * HD_;
    for (int i = tid; i < G_ * HD_; i += 256) qlds[i] = qbase[i];
    __syncthreads();

    const float* kbase = cache_k + ((size_t)b * NKV_ + kv) * SW_ * HD_;
    const float* vbase = cache_v + ((size_t)b * NKV_ + kv) * SW_ * HD_;
    const float* krowN = knew + ((size_t)b * NKV_ + kv) * HD_;
    const float* vrowN = vnew + ((size_t)b * NKV_ + kv) * HD_;

    float m[G_], l[G_], acc[G_][4];
    #pragma unroll
    for (int h = 0; h < G_; ++h) {
        m[h] = -1e30f; l[h] = 0.0f;
        acc[h][0] = acc[h][1] = acc[h][2] = acc[h][3] = 0.0f;
    }

    const int base = chunk * 128;
    for (int t = 0; t < 4; ++t) {
        const int s = base + t * 32 + lane;
        const float* kr = (s == slot) ? krowN : (kbase + (size_t)s * HD_);
        if (t < 3) __builtin_prefetch(kbase + (size_t)(s + 32) * HD_);

        float sc[G_];
        #pragma unroll
        for (int h = 0; h < G_; ++h) sc[h] = 0.0f;
        #pragma unroll 8
        for (int j = 0; j < HD_; j += 4) {
            f4 kk = __builtin_nontemporal_load((const f4*)(kr + j));
            #pragma unroll
            for (int h = 0; h < G_; ++h) {
                f4 qq = *(const f4*)(qlds + h * HD_ + j);
                sc[h] += qq.x * kk.x + qq.y * kk.y + qq.z * kk.z + qq.w * kk.w;
            }
        }

        const bool valid = (s <= cur);
        float p[G_];
        #pragma unroll
        for (int h = 0; h < G_; ++h) {
            const float sh = valid ? sc[h] : -1e9f;     // matches reference mask
            const float nm = fmaxf(m[h], waveMax(sh));
            const float corr = __expf(m[h] - nm);
            const float ph = __expf(sh - nm);
            l[h] = l[h] * corr + waveSum(ph);
            m[h] = nm;
            acc[h][0] *= corr; acc[h][1] *= corr; acc[h][2] *= corr; acc[h][3] *= corr;
            p[h] = ph;
        }

        const int sb = base + t * 32;
        for (int i = 0; i < 32; ++i) {
            const int si = sb + i;
            const float* vr = (si == slot) ? vrowN : (vbase + (size_t)si * HD_);
            f4 vv = __builtin_nontemporal_load((const f4*)(vr + 4 * lane));
            #pragma unroll
            for (int h = 0; h < G_; ++h) {
                const float ph = __shfl(p[h], i, 32);
                acc[h][0] += ph * vv.x; acc[h][1] += ph * vv.y;
                acc[h][2] += ph * vv.z; acc[h][3] += ph * vv.w;
            }
        }
    }

    const size_t pbase = (size_t)(((b * NKV_ + kv) * 32 + chunk) * G_) * PSTR_;
    #pragma unroll
    for (int h = 0; h < G_; ++h) {
        float* pp = part + pbase + (size_t)h * PSTR_;
        if (lane == 0) { pp[0] = m[h]; pp[1] = l[h]; }
        f4 a; a.x = acc[h][0]; a.y = acc[h][1]; a.z = acc[h][2]; a.w = acc[h][3];
        *(f4*)(pp + 4 + 4 * lane) = a;
    }
}

// ---------------------------------------------------------------------------
// Merge 32 chunk partials per (b, head) via log-sum-exp; one wave per head.
// ---------------------------------------------------------------------------
__global__ __launch_bounds__(256) void attn_combine(
    const float* __restrict__ part, float* __restrict__ ctx)
{
    const int lane = threadIdx.x & 31;
    const int gw = blockIdx.x * 8 + (threadIdx.x >> 5);
    if (gw >= B_ * NH_) return;
    const int b = gw / NH_, h = gw % NH_;
    const int kv = h / G_, g = h % G_;
    const float* p0 = part + (size_t)(((b * NKV_ + kv) * 32) * G_ + g) * PSTR_;
    const int cstr = G_ * PSTR_;

    float M = -1e30f;
    for (int c = 0; c < 32; ++c) M = fmaxf(M, p0[(size_t)c * cstr]);
    float L = 0.0f;
    for (int c = 0; c < 32; ++c)
        L += p0[(size_t)c * cstr + 1] * __expf(p0[(size_t)c * cstr] - M);

    f4 s; s.x = s.y = s.z = s.w = 0.0f;
    for (int c = 0; c < 32; ++c) {
        const float* pc = p0 + (size_t)c * cstr;
        const float w = __expf(pc[0] - M);
        f4 a = *(const f4*)(pc + 4 + 4 * lane);
        s.x += w * a.x; s.y += w * a.y; s.z += w * a.z; s.w += w * a.w;
    }
    const float inv = 1.0f / L;
    s.x *= inv; s.y *= inv; s.z *= inv; s.w *= inv;
    *(f4*)(ctx + (size_t)b * D_ + h * HD_ + 4 * lane) = s;
}

// final reduce of out-projection K-chunk partials into d_out
__global__ __launch_bounds__(256) void reduce_out(
    const float* __restrict__ part, float* __restrict__ out)
{
    const int i = blockIdx.x * 256 + threadIdx.x;
    if (i < B_ * D_)
        out[i] = part[i] + part[B_ * D_ + i] + part[2 * B_ * D_ + i] + part[3 * B_ * D_ + i];
}

extern "C" void kernel_launch(void* const* d_in, const int* in_sizes, int n_in,
                              void* d_out, int out_size, void* d_ws, size_t ws_size,
                              hipStream_t stream) {
    const float* x    = (const float*)d_in[0];
    const float* wqkv = (const float*)d_in[1];
    const float* bias = (const float*)d_in[2];
    const float* wo   = (const float*)d_in[3];
    const float* rot  = (const float*)d_in[4];
    const float* ck   = (const float*)d_in[5];
    const float* cv   = (const float*)d_in[6];
    const int*   cur  = (const int*)d_in[7];
    float* out = (float*)d_out;
    float* ws  = (float*)d_ws;

    // 1) QKV projection partials: 288 N-tiles * 4 K-chunks = 1152 waves
    gemm32_f32_wmma<<<(NQKV_/16)*KCH_/8, 256, 0, stream>>>(x, wqkv, ws + OFF_QKVP, D_, NQKV_);
    // 2) reduce + bias + RoPE + q-scale
    rope_assemble<<<(B_*NH_*64 + B_*NKV_*64 + B_*NKV_*HD_ + 255)/256, 256, 0, stream>>>(
        ws + OFF_QKVP, bias, rot, ws + OFF_QROT, ws + OFF_KNEW, ws + OFF_VNEW);
    // 3) flash-decode partials: 4096 waves
    attn_partial<<<B_*NKV_*32/8, 256, 0, stream>>>(
        ck, cv, ws + OFF_QROT, ws + OFF_KNEW, ws + OFF_VNEW, cur, ws + OFF_ATT);
    // 4) LSE merge: 896 waves
    attn_combine<<<B_*NH_/8, 256, 0, stream>>>(ws + OFF_ATT, ws + OFF_CTX);
    // 5) output projection partials: 224 N-tiles * 4 K-chunks = 896 waves
    gemm32_f32_wmma<<<(D_/16)*KCH_/8, 256, 0, stream>>>(ws + OFF_CTX, wo, ws + OFF_OUTP, D_, D_);
    // 6) final reduce into d_out
    reduce_out<<<(B_*D_ + 255)/256, 256, 0, stream>>>(ws + OFF_OUTP, out);
}